// RNNCapsule_87857851007023
// MI455X (gfx1250) — compile-verified
//
#include <hip/hip_runtime.h>
#include <hip/hip_bf16.h>

typedef __attribute__((ext_vector_type(2))) float v2f;
typedef __attribute__((ext_vector_type(8))) float v8f;

#define B_  32
#define N_  4096
#define D_  512
#define C_  5

// ---------------------------------------------------------------------------
// K1: e[b,n,c] = sum_d x[b,n,d] * W[d,c]   via V_WMMA_F32_16X16X4_F32
// block = 256 threads (8 waves). Each wave: one 16(n) x 16(c) tile, K=512.
// grid.x = B * (N/128)
// ---------------------------------------------------------------------------
__global__ void __launch_bounds__(256)
k1_logits(const float* __restrict__ x, const float* __restrict__ W,
          float* __restrict__ e) {
    __shared__ float WT[16 * 514];   // transposed, padded W: WT[c][d], stride 514

    const int nchunks = N_ / 128;
    const int b     = blockIdx.x / nchunks;
    const int chunk = blockIdx.x % nchunks;

    // Build WT: rows c<5 hold W[d][c], rows 5..15 are zero (their output
    // columns are discarded anyway, but keep them clean).
    for (int i = threadIdx.x; i < 16 * 512; i += 256) {
        int c = i >> 9;          // 0..15
        int d = i & 511;         // 0..511
        WT[c * 514 + d] = (c < C_) ? W[d * C_ + c] : 0.0f;
    }
    __syncthreads();

    const int wave  = threadIdx.x >> 5;
    const int lane  = threadIdx.x & 31;
    const int row   = lane & 15;     // M index (seq row) / N index (caps col)
    const int khalf = lane >> 4;     // K pairing: 0 -> K{0,1}, 1 -> K{2,3}

    const int n0 = chunk * 128 + wave * 16;

    const float* xp = x + ((size_t)b * N_ + n0 + row) * D_ + 2 * khalf;
    const float* wp = &WT[row * 514 + 2 * khalf];

    v8f acc = {};
    #pragma unroll 4
    for (int k = 0; k < D_; k += 4) {
        v2f a = *(const v2f*)(xp + k);   // A[m=row][k'], k' = v + 2*khalf
        v2f bb = *(const v2f*)(wp + k);  // B[k'][n=row]
        acc = __builtin_amdgcn_wmma_f32_16x16x4_f32(
            false, a, false, bb, (short)0, acc, false, false);
    }

    const int c = row;                   // output column = capsule
    if (c < C_) {
        #pragma unroll
        for (int r = 0; r < 8; ++r) {
            int n = n0 + r + 8 * khalf;  // output row = sequence position
            e[((size_t)b * N_ + n) * C_ + c] = acc[r];
        }
    }
}

// ---------------------------------------------------------------------------
// K2: per (b,c): softmax over n. Writes alphaT[b][c][n] (c-major, 16 rows).
// grid.x = B*C, block = 256
// ---------------------------------------------------------------------------
__global__ void __launch_bounds__(256)
k2_softmax(const float* __restrict__ e, float* __restrict__ alphaT) {
    __shared__ float red[256];
    const int b = blockIdx.x / C_;
    const int c = blockIdx.x % C_;
    const int tid = threadIdx.x;

    const float* ebc = e + (size_t)b * N_ * C_ + c;

    // --- max over n ---
    float m = -1e30f;
    for (int n = tid; n < N_; n += 256)
        m = fmaxf(m, ebc[(size_t)n * C_]);
    red[tid] = m;
    __syncthreads();
    for (int s = 128; s > 0; s >>= 1) {
        if (tid < s) red[tid] = fmaxf(red[tid], red[tid + s]);
        __syncthreads();
    }
    m = red[0];
    __syncthreads();

    // --- sum of exp ---
    float sl = 0.0f;
    for (int n = tid; n < N_; n += 256)
        sl += expf(ebc[(size_t)n * C_] - m);
    red[tid] = sl;
    __syncthreads();
    for (int s = 128; s > 0; s >>= 1) {
        if (tid < s) red[tid] += red[tid + s];
        __syncthreads();
    }
    const float inv = 1.0f / red[0];

    float* arow = alphaT + ((size_t)b * 16 + c) * N_;
    for (int n = tid; n < N_; n += 256)
        arow[n] = expf(ebc[(size_t)n * C_] - m) * inv;
}

// ---------------------------------------------------------------------------
// K3: v_c[b,c,d] = sum_n alphaT[b,c,n] * x[b,n,d]  via WMMA f32 16x16x4.
// Wave: 16(c, 5 live) x 16(d) tile, K = N = 4096 (1024 WMMA steps).
// grid.x = B * (D/128), block = 256 (8 waves, 16 d-cols each)
// ---------------------------------------------------------------------------
__global__ void __launch_bounds__(256)
k3_vc(const float* __restrict__ x, const float* __restrict__ alphaT,
      float* __restrict__ vc) {
    const int dchunks = D_ / 128;
    const int b      = blockIdx.x / dchunks;
    const int dchunk = blockIdx.x % dchunks;

    const int wave  = threadIdx.x >> 5;
    const int lane  = threadIdx.x & 31;
    const int m     = lane & 15;     // A row (capsule) and B col (d)
    const int khalf = lane >> 4;

    const int d0 = dchunk * 128 + wave * 16;

    // A[m][k'] = alphaT[b][m][nk + v + 2*khalf]  (contiguous float2)
    const float* ap = alphaT + ((size_t)b * 16 + m) * N_ + 2 * khalf;
    // B[k'][n'] = x[b][nk + v + 2*khalf][d0 + m]
    const float* xp = x + ((size_t)b * N_ + 2 * khalf) * D_ + d0 + m;

    v8f acc = {};
    #pragma unroll 2
    for (int nk = 0; nk < N_; nk += 4) {
        v2f a = *(const v2f*)(ap + nk);
        v2f bb;
        bb.x = xp[(size_t)nk * D_];
        bb.y = xp[(size_t)(nk + 1) * D_];
        acc = __builtin_amdgcn_wmma_f32_16x16x4_f32(
            false, a, false, bb, (short)0, acc, false, false);
    }

    const int d = d0 + (lane & 15);
    #pragma unroll
    for (int r = 0; r < 8; ++r) {
        int c = r + 8 * khalf;           // output row = capsule
        if (c < C_)
            vc[((size_t)b * C_ + c) * D_ + d] = acc[r];
    }
}

// ---------------------------------------------------------------------------
// K4: p = tanh(v_c . lin_w + lin_b); r_s = p * v_c
// grid.x = B*C, block = 256 (each thread owns 2 of the 512 d's)
// ---------------------------------------------------------------------------
__global__ void __launch_bounds__(256)
k4_gate(const float* __restrict__ vc, const float* __restrict__ lw,
        const float* __restrict__ lb, float* __restrict__ p_out,
        float* __restrict__ rs_out) {
    __shared__ float red[256];
    const int b = blockIdx.x / C_;
    const int c = blockIdx.x % C_;
    const int tid = threadIdx.x;

    const float* v = vc + ((size_t)b * C_ + c) * D_;
    float v0 = v[tid];
    float v1 = v[tid + 256];
    red[tid] = v0 * lw[tid] + v1 * lw[tid + 256];
    __syncthreads();
    for (int s = 128; s > 0; s >>= 1) {
        if (tid < s) red[tid] += red[tid + s];
        __syncthreads();
    }
    const float p = tanhf(red[0] + lb[0]);

    if (tid == 0) p_out[b * C_ + c] = p;
    float* rs = rs_out + ((size_t)b * C_ + c) * D_;
    rs[tid]       = p * v0;
    rs[tid + 256] = p * v1;
}

// ---------------------------------------------------------------------------
extern "C" void kernel_launch(void* const* d_in, const int* in_sizes, int n_in,
                              void* d_out, int out_size, void* d_ws, size_t ws_size,
                              hipStream_t stream) {
    const float* x  = (const float*)d_in[0];  // (B,N,D)
    const float* W  = (const float*)d_in[1];  // (D,C)
    const float* lw = (const float*)d_in[2];  // (D,1)
    const float* lb = (const float*)d_in[3];  // (1,)

    float* out = (float*)d_out;
    float* p_out  = out;                                   // B*C
    float* vc_out = out + (size_t)B_ * C_;                 // B*C*D
    float* rs_out = out + (size_t)B_ * C_ + (size_t)B_ * C_ * D_;

    float* ws      = (float*)d_ws;
    float* e_ws     = ws;                                  // B*N*C floats
    float* alphaT_ws = ws + (size_t)B_ * N_ * C_;          // B*16*N floats

    // K1: logits
    k1_logits<<<B_ * (N_ / 128), 256, 0, stream>>>(x, W, e_ws);
    // K2: softmax over n, emit transposed alpha
    k2_softmax<<<B_ * C_, 256, 0, stream>>>(e_ws, alphaT_ws);
    // K3: v_c
    k3_vc<<<B_ * (D_ / 128), 256, 0, stream>>>(x, alphaT_ws, vc_out);
    // K4: p and r_s
    k4_gate<<<B_ * C_, 256, 0, stream>>>(vc_out, lw, lb, p_out, rs_out);
}